// GCNLayer_37220186587487
// MI455X (gfx1250) — compile-verified
//
#include <hip/hip_runtime.h>
#include <stdint.h>

typedef float v2f __attribute__((ext_vector_type(2)));
typedef float v8f __attribute__((ext_vector_type(8)));

#define DIM 128
#define LDS_STRIDE 132   // 132 % 64 = 4 -> 16 tile rows hit 16 distinct banks

// ---- degree / normalization ------------------------------------------------

__global__ void gcn_deg_init(float* __restrict__ deg, int n) {
    int i = blockIdx.x * blockDim.x + threadIdx.x;
    if (i < n) deg[i] = 1.0f;                      // self-loop
}

__global__ void gcn_deg_accum(const int* __restrict__ dst, float* __restrict__ deg, int e) {
    int i = blockIdx.x * blockDim.x + threadIdx.x;
    if (i < e) atomicAdd(&deg[dst[i]], 1.0f);
}

__global__ void gcn_dinv(float* __restrict__ deg, int n) {
    int i = blockIdx.x * blockDim.x + threadIdx.x;
    if (i < n) deg[i] = rsqrtf(deg[i]);            // deg >= 1 always (self-loop)
}

// ---- GEMM: y = (x @ W^T) * dinv[row];  out = b + y * dinv[row] (self-loop) --
// Block = 16 rows x 128 cols. 8 waves; wave w computes the 16x16 tile at
// columns [16w, 16w+16). K=128 via 32x v_wmma_f32_16x16x4_f32 (exact f32).

__global__ void __launch_bounds__(256)
gcn_gemm_wmma(const float* __restrict__ x, const float* __restrict__ W,
              const float* __restrict__ bias, const float* __restrict__ dinv,
              float* __restrict__ y, float* __restrict__ out, int n) {
    __shared__ float lds[16 * LDS_STRIDE];
    __shared__ float sdinv[16];
    const int tid = threadIdx.x;
    const int rowBase = blockIdx.x * 16;

    // Stage 16x128 f32 tile of x into LDS (coalesced b128 loads).
    for (int t = tid; t < 512; t += 256) {
        int r  = t >> 5;          // 0..15 row in tile
        int c4 = t & 31;          // 0..31 float4 within row
        int row = rowBase + r;
        if (row >= n) row = n - 1;
        const float4 v = *(const float4*)(x + (size_t)row * DIM + c4 * 4);
        *(float4*)(&lds[r * LDS_STRIDE + c4 * 4]) = v;
    }
    // Stage the 16 dinv values for this row tile.
    if (tid < 16) {
        int row = rowBase + tid;
        sdinv[tid] = (row < n) ? dinv[row] : 0.0f;
    }
    __syncthreads();

    const int wave = tid >> 5;
    const int lane = tid & 31;
    const int half = lane >> 4;         // K-pair select (ISA 16x4 f32 A layout)
    const int l16  = lane & 15;
    const int ncol = wave * 16 + l16;   // output column == row of W

    const float* __restrict__ wrow = W + (size_t)ncol * DIM;
    const float* __restrict__ arow = &lds[l16 * LDS_STRIDE];

    v8f acc = {};
    #pragma unroll 4
    for (int kb = 0; kb < DIM; kb += 4) {
        const int kk = kb + 2 * half;
        v2f a; a.x = arow[kk]; a.y = arow[kk + 1];   // A[M=l16][kk..kk+1]
        v2f b; b.x = wrow[kk]; b.y = wrow[kk + 1];   // B[kk..kk+1][N=l16] = W[ncol][kk..]
        acc = __builtin_amdgcn_wmma_f32_16x16x4_f32(
                  false, a, false, b, (short)0, acc, false, false);
    }

    // C/D layout: VGPR j, lanes<16 -> M=j, lanes>=16 -> M=j+8; N = lane%16.
    const float bcol = bias[ncol];
    if (rowBase + 16 <= n) {
        // Fast path: whole tile in range, no per-element predication.
        #pragma unroll
        for (int j = 0; j < 8; ++j) {
            const int   r   = j + 8 * half;
            const float di  = sdinv[r];
            const float yv  = acc[j] * di;
            const size_t off = (size_t)(rowBase + r) * DIM + ncol;
            y[off]   = yv;
            out[off] = bcol + yv * di;               // bias + self-loop msg
        }
    } else {
        #pragma unroll
        for (int j = 0; j < 8; ++j) {
            const int r   = j + 8 * half;
            const int row = rowBase + r;
            if (row < n) {
                const float di = sdinv[r];
                const float yv = acc[j] * di;
                const size_t off = (size_t)row * DIM + ncol;
                y[off]   = yv;
                out[off] = bcol + yv * di;
            }
        }
    }
}

// ---- Edge scatter: out[dst] += y[src] * dinv[dst] --------------------------
// One wave32 per edge: each lane moves one float4 (128 floats/row) via a
// coalesced global_load_b128 and 4 dword global_atomic_add_f32. y/out/dinv
// (~103 MB) stay resident in the 192 MB L2, so this runs at L2 rate.

__global__ void __launch_bounds__(256)
gcn_edge_scatter(const int* __restrict__ src, const int* __restrict__ dst,
                 const float* __restrict__ y, const float* __restrict__ dinv,
                 float* __restrict__ out, int e) {
    const int gid  = blockIdx.x * blockDim.x + threadIdx.x;
    const int edge = gid >> 5;
    const int lane = gid & 31;
    if (edge >= e) return;

    const int s = src[edge];
    const int d = dst[edge];
    const float w = dinv[d];

    const float4 v = *(const float4*)(y + (size_t)s * DIM + lane * 4);
    float* o = out + (size_t)d * DIM + lane * 4;
    atomicAdd(o + 0, v.x * w);
    atomicAdd(o + 1, v.y * w);
    atomicAdd(o + 2, v.z * w);
    atomicAdd(o + 3, v.w * w);
}

// ---- launch ----------------------------------------------------------------

extern "C" void kernel_launch(void* const* d_in, const int* in_sizes, int n_in,
                              void* d_out, int out_size, void* d_ws, size_t ws_size,
                              hipStream_t stream) {
    const float* x    = (const float*)d_in[0];
    const int*   ei   = (const int*)d_in[1];   // [2, E]: row0 = src, row1 = dst
    const float* W    = (const float*)d_in[2];
    const float* bias = (const float*)d_in[3];
    float* out = (float*)d_out;

    const int N = in_sizes[0] / DIM;
    const int E = in_sizes[1] / 2;

    float* dinv = (float*)d_ws;      // N floats (deg, then rsqrt in place)
    float* y    = dinv + N;          // N*DIM floats

    const int T = 256;
    gcn_deg_init <<<(N + T - 1) / T, T, 0, stream>>>(dinv, N);
    gcn_deg_accum<<<(E + T - 1) / T, T, 0, stream>>>(ei + E, dinv, E);
    gcn_dinv     <<<(N + T - 1) / T, T, 0, stream>>>(dinv, N);

    gcn_gemm_wmma<<<(N + 15) / 16, T, 0, stream>>>(x, W, bias, dinv, y, out, N);

    const long long scatterThreads = (long long)E * 32;
    gcn_edge_scatter<<<(unsigned)((scatterThreads + T - 1) / T), T, 0, stream>>>(
        ei, ei + E, y, dinv, out, E);
}